// ChebyNet_36627481101156
// MI455X (gfx1250) — compile-verified
//
#include <hip/hip_runtime.h>
#include <math.h>

#define Nn   100000
#define Ee   1600000
#define FIN  128
#define FHID 64
#define FOUT 16
#define MTILES (Nn / 16)   // 6250, exact

typedef __attribute__((ext_vector_type(2))) float v2f;
typedef __attribute__((ext_vector_type(8))) float v8f;

static_assert(Nn % 16 == 0, "M must tile by 16");

// Map W[k][c] (row-major, ncols = 16*nct) into the WMMA B-fragment LDS layout:
// fragment (k4 = k/4, nt = c/16) stored as 64 floats, lane-major: lane*2 + j,
// where lane = (k%4)/2 * 16 + (c%16), j = k%2.  Reads are then one aligned
// float2 per lane -> every LDS bank hit exactly once (conflict-free).
__device__ __forceinline__ int frag_index(int k, int c, int nct) {
  int k4 = k >> 2, kr = k & 3;
  int hl = kr >> 1, j = kr & 1;
  int nt = c >> 4, r = c & 15;
  int lane = (hl << 4) | r;
  return (((k4 * nct + nt) << 6) + (lane << 1) + j);
}

// ---------------- degree / norm ----------------
__global__ void degree_k(const int* __restrict__ src, const int* __restrict__ dst,
                         float* __restrict__ deg) {
  int e = blockIdx.x * blockDim.x + threadIdx.x;
  if (e >= Ee) return;
  int s = src[e];
  if (s != dst[e]) atomicAdd(&deg[s], 1.0f);
}

__global__ void dinv_k(const float* __restrict__ deg, float* __restrict__ dinv) {
  int i = blockIdx.x * blockDim.x + threadIdx.x;
  if (i >= Nn) return;
  float d = deg[i];
  dinv[i] = d > 0.0f ? rsqrtf(d) : 0.0f;
}

// ---------------- layer-1 dual GEMM: Z0 = X@W0, Z1 = X@W1 (M=N nodes, K=128, N=64)
__global__ void __launch_bounds__(256) gemm1_k(const float* __restrict__ X,
                                               const float* __restrict__ W0,
                                               const float* __restrict__ W1,
                                               float* __restrict__ Z0,
                                               float* __restrict__ Z1) {
  __shared__ float sB0[FIN / 4 * 4 * 64];  // 8192 floats = 32 KB
  __shared__ float sB1[FIN / 4 * 4 * 64];  // 8192 floats = 32 KB
  for (int i = threadIdx.x; i < FIN * FHID; i += blockDim.x) {
    int k = i >> 6, c = i & 63;  // W row-major [128][64]
    int fi = frag_index(k, c, 4);
    sB0[fi] = W0[i];
    sB1[fi] = W1[i];
  }
  __syncthreads();

  int wv = threadIdx.x >> 5;
  int lane = threadIdx.x & 31;
  int hl = lane >> 4, r = lane & 15;
  int mt = blockIdx.x * 8 + wv;
  if (mt >= MTILES) return;
  int m0 = mt << 4;

  v8f acc0[4] = {};
  v8f acc1[4] = {};
  const float* xrow = X + (size_t)(m0 + r) * FIN;
  const v2f* B0 = (const v2f*)sB0;
  const v2f* B1 = (const v2f*)sB1;

#pragma unroll 4
  for (int k4 = 0; k4 < FIN / 4; ++k4) {
    int kb = (k4 << 2) + (hl << 1);
    v2f a = *(const v2f*)(xrow + kb);  // A frag: lane holds row m0+r, K = kb..kb+1
#pragma unroll
    for (int nt = 0; nt < 4; ++nt) {
      v2f b0 = B0[(((k4 << 2) + nt) << 5) + lane];
      v2f b1 = B1[(((k4 << 2) + nt) << 5) + lane];
      acc0[nt] = __builtin_amdgcn_wmma_f32_16x16x4_f32(false, a, false, b0,
                                                       (short)0, acc0[nt], false, false);
      acc1[nt] = __builtin_amdgcn_wmma_f32_16x16x4_f32(false, a, false, b1,
                                                       (short)0, acc1[nt], false, false);
    }
  }

#pragma unroll
  for (int nt = 0; nt < 4; ++nt) {
#pragma unroll
    for (int v = 0; v < 8; ++v) {
      int row = m0 + v + (hl << 3);
      int col = (nt << 4) + r;
      Z0[(size_t)row * FHID + col] = acc0[nt][v];
      Z1[(size_t)row * FHID + col] = acc1[nt][v];
    }
  }
}

// ---------------- edge scatter in HID space: Tx1[dst] += norm * Z1[src] -----
__global__ void __launch_bounds__(256) aggregate1_k(const int* __restrict__ src,
                                                    const int* __restrict__ dst,
                                                    const float* __restrict__ dinv,
                                                    const float* __restrict__ Z1,
                                                    float* __restrict__ Tx1) {
  int wid = (blockIdx.x * blockDim.x + threadIdx.x) >> 5;  // one wave per edge
  int lane = threadIdx.x & 31;
  if (wid >= Ee) return;
  int s = src[wid], d = dst[wid];
  if (s == d) return;  // self-loops removed (w = 0)
  float nrm = -dinv[s] * dinv[d];
  const float2 z = *(const float2*)(Z1 + (size_t)s * FHID + lane * 2);
  float* p = Tx1 + (size_t)d * FHID + lane * 2;
  atomicAdd(p, nrm * z.x);
  atomicAdd(p + 1, nrm * z.y);
}

// ---------------- h = relu(Z0 + Tx1 + b1), in place over Z0 ----------------
__global__ void bias_relu_k(float* __restrict__ Z0, const float* __restrict__ Tx1,
                            const float* __restrict__ b1) {
  size_t base = ((size_t)blockIdx.x * blockDim.x + threadIdx.x) * 4;
  if (base >= (size_t)Nn * FHID) return;
  float4 z = *(const float4*)(Z0 + base);
  float4 t = *(const float4*)(Tx1 + base);
  int f = (int)(base & (FHID - 1));
  float4 b = *(const float4*)(b1 + f);
  float4 o;
  o.x = fmaxf(z.x + t.x + b.x, 0.0f);
  o.y = fmaxf(z.y + t.y + b.y, 0.0f);
  o.z = fmaxf(z.z + t.z + b.z, 0.0f);
  o.w = fmaxf(z.w + t.w + b.w, 0.0f);
  *(float4*)(Z0 + base) = o;
}

// ---------------- layer-2 dual GEMM: Y0 = H@W0, Y1 = H@W1 (K=64, N=16) -----
__global__ void __launch_bounds__(256) gemm2_k(const float* __restrict__ H,
                                               const float* __restrict__ W0,
                                               const float* __restrict__ W1,
                                               float* __restrict__ Y0,
                                               float* __restrict__ Y1) {
  __shared__ float sB0[FHID / 4 * 64];  // 1024 floats
  __shared__ float sB1[FHID / 4 * 64];
  for (int i = threadIdx.x; i < FHID * FOUT; i += blockDim.x) {
    int k = i >> 4, c = i & 15;  // W row-major [64][16]
    int fi = frag_index(k, c, 1);
    sB0[fi] = W0[i];
    sB1[fi] = W1[i];
  }
  __syncthreads();

  int wv = threadIdx.x >> 5;
  int lane = threadIdx.x & 31;
  int hl = lane >> 4, r = lane & 15;
  int mt = blockIdx.x * 8 + wv;
  if (mt >= MTILES) return;
  int m0 = mt << 4;

  v8f a0 = {};
  v8f a1 = {};
  const float* hrow = H + (size_t)(m0 + r) * FHID;
  const v2f* B0 = (const v2f*)sB0;
  const v2f* B1 = (const v2f*)sB1;

#pragma unroll
  for (int k4 = 0; k4 < FHID / 4; ++k4) {
    int kb = (k4 << 2) + (hl << 1);
    v2f a = *(const v2f*)(hrow + kb);
    v2f b0 = B0[(k4 << 5) + lane];
    v2f b1 = B1[(k4 << 5) + lane];
    a0 = __builtin_amdgcn_wmma_f32_16x16x4_f32(false, a, false, b0, (short)0, a0, false, false);
    a1 = __builtin_amdgcn_wmma_f32_16x16x4_f32(false, a, false, b1, (short)0, a1, false, false);
  }

#pragma unroll
  for (int v = 0; v < 8; ++v) {
    int row = m0 + v + (hl << 3);
    Y0[(size_t)row * FOUT + r] = a0[v];
    Y1[(size_t)row * FOUT + r] = a1[v];
  }
}

// ---------------- edge scatter in OUT space: Ty[dst] += norm * Y1[src] -----
__global__ void __launch_bounds__(256) aggregate2_k(const int* __restrict__ src,
                                                    const int* __restrict__ dst,
                                                    const float* __restrict__ dinv,
                                                    const float* __restrict__ Y1,
                                                    float* __restrict__ Ty) {
  long long t = (long long)blockIdx.x * blockDim.x + threadIdx.x;
  if (t >= (long long)Ee * FOUT) return;
  int e = (int)(t >> 4), f = (int)(t & 15);
  int s = src[e], d = dst[e];
  if (s == d) return;
  float nrm = -dinv[s] * dinv[d];
  atomicAdd(&Ty[(size_t)d * FOUT + f], nrm * Y1[(size_t)s * FOUT + f]);
}

// ---------------- out = log_softmax(Y0 + Ty + b2) --------------------------
__global__ void final_k(const float* __restrict__ Y0, const float* __restrict__ Ty,
                        const float* __restrict__ b2, float* __restrict__ out) {
  int i = blockIdx.x * blockDim.x + threadIdx.x;
  if (i >= Nn) return;
  float o[FOUT];
  float m = -INFINITY;
#pragma unroll
  for (int f = 0; f < FOUT; ++f) {
    o[f] = Y0[(size_t)i * FOUT + f] + Ty[(size_t)i * FOUT + f] + b2[f];
    m = fmaxf(m, o[f]);
  }
  float s = 0.0f;
#pragma unroll
  for (int f = 0; f < FOUT; ++f) s += expf(o[f] - m);
  float lse = m + logf(s);
#pragma unroll
  for (int f = 0; f < FOUT; ++f) out[(size_t)i * FOUT + f] = o[f] - lse;
}

extern "C" void kernel_launch(void* const* d_in, const int* in_sizes, int n_in,
                              void* d_out, int out_size, void* d_ws, size_t ws_size,
                              hipStream_t stream) {
  const float* x    = (const float*)d_in[0];
  const int*   ei   = (const int*)d_in[1];
  const int*   srcv = ei;        // edge_index[0]
  const int*   dstv = ei + Ee;   // edge_index[1]
  const float* W0_1 = (const float*)d_in[2];
  const float* W1_1 = (const float*)d_in[3];
  const float* b1   = (const float*)d_in[4];
  const float* W0_2 = (const float*)d_in[5];
  const float* W1_2 = (const float*)d_in[6];
  const float* b2   = (const float*)d_in[7];
  float* out = (float*)d_out;

  // Workspace layout (floats): Z0|h (6.4M) | Z1 (6.4M, reused for Y0/Y1) |
  // Tx1 (6.4M, reused for Ty) | deg (100k) | dinv (100k)   ~= 78 MB
  float* ws   = (float*)d_ws;
  float* Z0   = ws;                         // N*HID; becomes h in place
  float* Z1   = Z0 + (size_t)Nn * FHID;
  float* Tx1  = Z1 + (size_t)Nn * FHID;
  float* deg  = Tx1 + (size_t)Nn * FHID;
  float* dinv = deg + Nn;
  float* Y0 = Z1;                           // Z1 dead after aggregate1
  float* Y1 = Z1 + (size_t)Nn * FOUT;
  float* Ty = Tx1;                          // Tx1 dead after bias_relu

  hipMemsetAsync(deg, 0, Nn * sizeof(float), stream);
  hipMemsetAsync(Tx1, 0, (size_t)Nn * FHID * sizeof(float), stream);

  degree_k<<<(Ee + 255) / 256, 256, 0, stream>>>(srcv, dstv, deg);
  dinv_k<<<(Nn + 255) / 256, 256, 0, stream>>>(deg, dinv);

  gemm1_k<<<(MTILES + 7) / 8, 256, 0, stream>>>(x, W0_1, W1_1, Z0, Z1);
  aggregate1_k<<<Ee / 8, 256, 0, stream>>>(srcv, dstv, dinv, Z1, Tx1);
  bias_relu_k<<<((size_t)Nn * FHID / 4 + 255) / 256, 256, 0, stream>>>(Z0, Tx1, b1);

  gemm2_k<<<(MTILES + 7) / 8, 256, 0, stream>>>(Z0, W0_2, W1_2, Y0, Y1);
  hipMemsetAsync(Ty, 0, (size_t)Nn * FOUT * sizeof(float), stream);
  aggregate2_k<<<(int)(((long long)Ee * FOUT + 255) / 256), 256, 0, stream>>>(
      srcv, dstv, dinv, Y1, Ty);

  final_k<<<(Nn + 255) / 256, 256, 0, stream>>>(Y0, Ty, b2, out);
}